// GPSConv_36404142801037
// MI455X (gfx1250) — compile-verified
//
#include <hip/hip_runtime.h>
#include <hip/hip_bf16.h>
#include <math.h>

#define N_NODES   65536
#define N_GRAPHS  64
#define MAX_LEN   1024
#define CH        128
#define D_STATE   16
#define D_CONV    4
#define DT_RANK   8
#define N_EDGES   1048576
#define BN_EPS    1e-5f
#define XPN       48   // x_proj output padded 40 -> 48 (3 WMMA N-tiles)
#define GEMM_KMAX 256  // largest K used (MLP second GEMM)

typedef __attribute__((ext_vector_type(2))) float v2f;
typedef __attribute__((ext_vector_type(8))) float v8f;
typedef __attribute__((ext_vector_type(4))) int   v4i;

#define AS1 __attribute__((address_space(1)))
#define AS3 __attribute__((address_space(3)))

#if defined(__has_builtin)
#  if __has_builtin(__builtin_amdgcn_global_load_async_to_lds_b128)
#    define HAVE_ASYNC_LDS 1
#  endif
#endif
#ifndef HAVE_ASYNC_LDS
#  define HAVE_ASYNC_LDS 0
#endif

__device__ __forceinline__ void wait_asynccnt0() {
#if defined(__has_builtin)
#  if __has_builtin(__builtin_amdgcn_s_wait_asynccnt)
  __builtin_amdgcn_s_wait_asynccnt(0);
  return;
#  endif
#endif
  asm volatile("s_wait_asynccnt 0x0" ::: "memory");
}

// ---------------------------------------------------------------------------
// fp32 WMMA GEMM: C[M x N] = act(A[M x K] @ B[K x N] + bias)
// Block = 256 threads (8 wave32) covering a 128x16 output tile: all 8 waves
// share one N-tile, so the B column slice (K x 16 f32, <= 16 KB) is staged in
// LDS once per block via GLOBAL_LOAD_ASYNC_TO_LDS_B128 (fallback: plain copy),
// then each wave runs V_WMMA_F32_16X16X4_F32 over K in steps of 4 with:
//   A fragment: one global_load_b64 per lane per step (adjacent k pair)
//   B fragment: two ds_load_b32 per lane per step
// M is a multiple of 128 and N a multiple of 16 for every call site, so EXEC
// stays all-ones for every wave executing WMMA.
// ---------------------------------------------------------------------------
template <int ACT> // 0 = none, 1 = relu
__global__ void __launch_bounds__(256)
gemm_wmma_f32(const float* __restrict__ A, const float* __restrict__ B,
              const float* __restrict__ bias, float* __restrict__ C,
              int M, int N, int K, int mblocks) {
  __shared__ float sB[GEMM_KMAX * 16];

  int bm = blockIdx.x % mblocks;       // which 128-row block
  int tn = blockIdx.x / mblocks;       // which 16-col tile
  int waveId = threadIdx.x >> 5;
  int lane   = threadIdx.x & 31;
  int tm = bm * 8 + waveId;

  // ---- stage B[:, tn*16 .. tn*16+16) into LDS (row-major [K][16]) ----
  int nchunk = K * 4;                  // float4 chunks (4 per 16-col row)
  for (int c = threadIdx.x; c < nchunk; c += 256) {
    int r = c >> 2;
    int p = (c & 3) << 2;
    const float* gsrc = B + (size_t)r * N + tn * 16 + p;
    float* ldst = &sB[r * 16 + p];
#if HAVE_ASYNC_LDS
    __builtin_amdgcn_global_load_async_to_lds_b128(
        (AS1 v4i*)(AS1 void*)gsrc, (AS3 v4i*)(AS3 void*)ldst, 0, 0);
#else
    *(float4*)ldst = *(const float4*)gsrc;
#endif
  }
#if HAVE_ASYNC_LDS
  wait_asynccnt0();
#endif
  __syncthreads();

  int half = lane >> 4;                // 0: k pair {0,1}, 1: k pair {2,3}
  int l15  = lane & 15;
  size_t row = (size_t)tm * 16 + l15;  // A row this lane supplies
  int col = tn * 16 + l15;             // B/C column this lane supplies
  const float2* Arow = (const float2*)(A + row * (size_t)K);

  v8f acc = {};
  for (int k = 0; k < K; k += 4) {
    float2 av = Arow[(k >> 1) + half];     // A[row][k+2*half], A[row][k+2*half+1]
    int ka = k + half * 2;
    v2f a, b;
    a.x = av.x;
    a.y = av.y;
    b.x = sB[ka * 16 + l15];
    b.y = sB[ka * 16 + 16 + l15];
    acc = __builtin_amdgcn_wmma_f32_16x16x4_f32(
        /*neg_a=*/false, a, /*neg_b=*/false, b,
        /*c_mod=*/(short)0, acc, /*reuse_a=*/false, /*reuse_b=*/false);
  }

  size_t rbase = (size_t)tm * 16 + half * 8;
  float badd = bias ? bias[col] : 0.0f;
#pragma unroll
  for (int r = 0; r < 8; ++r) {
    float v = acc[r] + badd;
    if (ACT == 1) v = fmaxf(v, 0.0f);
    C[(rbase + r) * (size_t)N + col] = v;
  }
}

// ---------------------------------------------------------------------------
// Elementwise / graph kernels
// ---------------------------------------------------------------------------
__global__ void k_fill(float* __restrict__ p, float v, int n) {
  int i = blockIdx.x * blockDim.x + threadIdx.x;
  if (i < n) p[i] = v;
}

__global__ void k_deg_count(const int* __restrict__ dst, float* __restrict__ deg) {
  int e = blockIdx.x * blockDim.x + threadIdx.x;
  if (e < N_EDGES) atomicAdd(&deg[dst[e]], 1.0f);
}

__global__ void k_rsqrt_inplace(float* __restrict__ p, int n) {
  int i = blockIdx.x * blockDim.x + threadIdx.x;
  if (i < n) p[i] = rsqrtf(p[i]);
}

// agg = xw * dinv^2 (self loop) + gcn_b + x (residual)
__global__ void k_agg_init(const float* __restrict__ xw, const float* __restrict__ x,
                           const float* __restrict__ dinv, const float* __restrict__ gb,
                           float* __restrict__ agg) {
  size_t i = (size_t)blockIdx.x * blockDim.x + threadIdx.x;
  int r = (int)(i >> 7), c = (int)(i & 127);
  float di = dinv[r];
  agg[i] = xw[i] * di * di + gb[c] + x[i];
}

// 32 threads per edge, 4 channels each: agg[dst] += xw[src] * dinv[src]*dinv[dst]
__global__ void k_edge_scatter(const int* __restrict__ ei, const float* __restrict__ xw,
                               const float* __restrict__ dinv, float* __restrict__ agg) {
  size_t gid = (size_t)blockIdx.x * blockDim.x + threadIdx.x;
  int e  = (int)(gid >> 5);
  int c0 = (int)(gid & 31) << 2;
  int s = ei[e], d = ei[N_EDGES + e];
  float nrm = dinv[s] * dinv[d];
  const float* xs = xw + (size_t)s * CH + c0;
  float* ad = agg + (size_t)d * CH + c0;
  atomicAdd(ad + 0, xs[0] * nrm);
  atomicAdd(ad + 1, xs[1] * nrm);
  atomicAdd(ad + 2, xs[2] * nrm);
  atomicAdd(ad + 3, xs[3] * nrm);
}

// Per-channel sum / sumsq over rows x CH  (128 threads per block, one channel each)
__global__ void __launch_bounds__(CH) k_bn_stats(const float* __restrict__ x,
                                                 float* __restrict__ stats, int rows) {
  int c = threadIdx.x;
  float s = 0.0f, s2 = 0.0f;
  for (int r = blockIdx.x; r < rows; r += gridDim.x) {
    float v = x[(size_t)r * CH + c];
    s += v; s2 += v * v;
  }
  atomicAdd(&stats[c], s);
  atomicAdd(&stats[CH + c], s2);
}

__global__ void k_bn_apply(const float* __restrict__ x, const float* __restrict__ stats,
                           const float* __restrict__ g, const float* __restrict__ b,
                           float* __restrict__ out, int rows) {
  size_t i = (size_t)blockIdx.x * blockDim.x + threadIdx.x;
  int c = (int)(i & 127);
  float inv = 1.0f / (float)rows;
  float mu  = stats[c] * inv;
  float var = stats[CH + c] * inv - mu * mu;
  out[i] = (x[i] - mu) * rsqrtf(var + BN_EPS) * g[c] + b[c];
}

// causal depthwise conv(4) + bias + SiLU;  x_in = xz[:, 0:128] (row stride 256)
__global__ void k_conv_silu(const float* __restrict__ xz, const float* __restrict__ cw,
                            const float* __restrict__ cb, float* __restrict__ xc) {
  size_t i = (size_t)blockIdx.x * blockDim.x + threadIdx.x;
  int row = (int)(i >> 7), c = (int)(i & 127);
  int t = row & (MAX_LEN - 1);
  float acc = cb[c];
#pragma unroll
  for (int k = 0; k < D_CONV; ++k) {
    int off = k - (D_CONV - 1);
    if (t + off >= 0)
      acc += xz[(size_t)(row + off) * (2 * CH) + c] * cw[c * D_CONV + k];
  }
  xc[i] = acc / (1.0f + expf(-acc));   // SiLU
}

__global__ void k_pad_xproj(const float* __restrict__ w, float* __restrict__ wp) {
  int i = blockIdx.x * blockDim.x + threadIdx.x;
  if (i >= CH * XPN) return;
  int r = i / XPN, c = i - r * XPN;
  wp[i] = (c < DT_RANK + 2 * D_STATE) ? w[r * (DT_RANK + 2 * D_STATE) + c] : 0.0f;
}

// dt = softplus(dbl[:, :8] @ dt_w + dt_b)
__global__ void k_dt(const float* __restrict__ dbl, const float* __restrict__ dtw,
                     const float* __restrict__ dtb, float* __restrict__ dt) {
  size_t i = (size_t)blockIdx.x * blockDim.x + threadIdx.x;
  int row = (int)(i >> 7), c = (int)(i & 127);
  const float* dr = dbl + (size_t)row * XPN;
  float acc = dtb[c];
#pragma unroll
  for (int r = 0; r < DT_RANK; ++r) acc += dr[r] * dtw[r * CH + c];
  dt[i] = (acc > 20.0f) ? acc : log1pf(expf(acc));
}

// Selective scan: one block per graph, one lane per channel; 16 states in regs.
// B_t / C_t (16 floats each) staged in LDS cooperatively each timestep.
// A is pre-scaled by log2(e) so the recurrence uses bare v_exp_f32 (base-2).
__global__ void __launch_bounds__(CH) k_scan(const float* __restrict__ dbl,
                                             const float* __restrict__ dt,
                                             const float* __restrict__ xc,
                                             const float* __restrict__ alog,
                                             float* __restrict__ y) {
  int g = blockIdx.x, c = threadIdx.x;
  const float LOG2E = 1.44269504088896340736f;
  float A2[D_STATE], h[D_STATE];
#pragma unroll
  for (int s = 0; s < D_STATE; ++s) {
    A2[s] = -expf(alog[c * D_STATE + s]) * LOG2E;
    h[s] = 0.0f;
  }
  __shared__ float sB[D_STATE], sC[D_STATE];
  for (int t = 0; t < MAX_LEN; ++t) {
    size_t row = (size_t)g * MAX_LEN + t;
    __syncthreads();
    if (c < 2 * D_STATE) {
      // c<16 fills sB (dbl cols 8..23), c in [16,32) fills sC (dbl cols 24..39)
      float v = dbl[row * XPN + DT_RANK + c];
      if (c < D_STATE) sB[c] = v; else sC[c - D_STATE] = v;
    }
    __syncthreads();
    float dv = dt[row * CH + c];
    float xv = xc[row * CH + c];
    float acc = 0.0f;
#pragma unroll
    for (int s = 0; s < D_STATE; ++s) {
      h[s] = exp2f(dv * A2[s]) * h[s] + dv * sB[s] * xv;
      acc += h[s] * sC[s];
    }
    y[row * CH + c] = acc;
  }
}

// y = (y + xc * Dp) * silu(z)   (z = xz[:, 128:256])
__global__ void k_gate(const float* __restrict__ xz, const float* __restrict__ xc,
                       const float* __restrict__ Dp, float* __restrict__ y) {
  size_t i = (size_t)blockIdx.x * blockDim.x + threadIdx.x;
  int row = (int)(i >> 7), c = (int)(i & 127);
  float z = xz[(size_t)row * (2 * CH) + CH + c];
  float v = y[i] + xc[i] * Dp[c];
  y[i] = v * (z / (1.0f + expf(-z)));
}

__global__ void k_add(const float* __restrict__ a, const float* __restrict__ b,
                      float* __restrict__ out, size_t n) {
  size_t i = (size_t)blockIdx.x * blockDim.x + threadIdx.x;
  if (i < n) out[i] = a[i] + b[i];
}

// ---------------------------------------------------------------------------
// Host-side orchestration
// ---------------------------------------------------------------------------
static inline void launch_gemm(const float* A, const float* B, const float* bias,
                               float* C, int M, int N, int K, int act,
                               hipStream_t stream) {
  int mblocks = M / 128;                 // 128 rows per block (8 waves x 16)
  int blocks  = mblocks * (N / 16);      // one 16-col tile per block
  if (act)
    gemm_wmma_f32<1><<<blocks, 256, 0, stream>>>(A, B, bias, C, M, N, K, mblocks);
  else
    gemm_wmma_f32<0><<<blocks, 256, 0, stream>>>(A, B, bias, C, M, N, K, mblocks);
}

static inline void run_bn(const float* in, float* stats, const float* g,
                          const float* b, float* out, hipStream_t stream) {
  k_fill<<<1, 256, 0, stream>>>(stats, 0.0f, 2 * CH);
  k_bn_stats<<<256, CH, 0, stream>>>(in, stats, N_NODES);
  k_bn_apply<<<(N_NODES * CH) / 256, 256, 0, stream>>>(in, stats, g, b, out, N_NODES);
}

extern "C" void kernel_launch(void* const* d_in, const int* in_sizes, int n_in,
                              void* d_out, int out_size, void* d_ws, size_t ws_size,
                              hipStream_t stream) {
  const float* x      = (const float*)d_in[0];
  const int*   ei     = (const int*)d_in[1];
  // d_in[2] (batch) is repeat(arange(64),1024): dense scatter == exact reshape.
  const float* gcn_w  = (const float*)d_in[3];
  const float* gcn_b  = (const float*)d_in[4];
  const float* inpw   = (const float*)d_in[5];
  const float* convw  = (const float*)d_in[6];
  const float* convb  = (const float*)d_in[7];
  const float* xprojw = (const float*)d_in[8];
  const float* dtw    = (const float*)d_in[9];
  const float* dtb    = (const float*)d_in[10];
  const float* alog   = (const float*)d_in[11];
  const float* Dp     = (const float*)d_in[12];
  const float* outpw  = (const float*)d_in[13];
  const float* w1     = (const float*)d_in[14];
  const float* b1     = (const float*)d_in[15];
  const float* w2     = (const float*)d_in[16];
  const float* b2     = (const float*)d_in[17];
  const float* bn1g   = (const float*)d_in[18];
  const float* bn1b   = (const float*)d_in[19];
  const float* bn2g   = (const float*)d_in[20];
  const float* bn2b   = (const float*)d_in[21];
  const float* bn3g   = (const float*)d_in[22];
  const float* bn3b   = (const float*)d_in[23];
  float* out = (float*)d_out;
  float* ws  = (float*)d_ws;

  const size_t NC = (size_t)N_NODES * CH;
  float* xw   = ws;                 // [N,128] gcn xw   -> later reused as out0
  float* agg  = ws + 1 * NC;        // [N,128] gcn agg  -> h1
  float* xz   = ws + 2 * NC;        // [N,256] in_proj  -> later MLP hidden
  float* xc   = ws + 4 * NC;        // [N,128] conv+silu-> later MLP out / final pre-BN
  float* dtb_ = ws + 5 * NC;        // [N,128] dt
  float* y    = ws + 6 * NC;        // [N,128] scan out -> gated
  float* mo   = ws + 7 * NC;        // [N,128] out_proj -> h2
  float* dbl  = ws + 8 * NC;        // [N,48]  x_proj out (padded)
  float* dinv = dbl + (size_t)N_NODES * XPN;  // [N] degree -> rsqrt
  float* xpw  = dinv + N_NODES;     // [128,48] padded x_proj_w
  float* stats = xpw + CH * XPN;    // [256] BN scratch

  const int EW = 256;
  const int ncBlocks = (int)(NC / EW);

  // --- GCN ---
  k_fill<<<N_NODES / EW, EW, 0, stream>>>(dinv, 1.0f, N_NODES);      // self loop
  k_deg_count<<<N_EDGES / EW, EW, 0, stream>>>(ei + N_EDGES, dinv);
  k_rsqrt_inplace<<<N_NODES / EW, EW, 0, stream>>>(dinv, N_NODES);
  launch_gemm(x, gcn_w, nullptr, xw, N_NODES, CH, CH, 0, stream);    // xw = x @ gcn_w
  k_agg_init<<<ncBlocks, EW, 0, stream>>>(xw, x, dinv, gcn_b, agg);
  k_edge_scatter<<<(int)(((size_t)N_EDGES * 32) / EW), EW, 0, stream>>>(ei, xw, dinv, agg);
  run_bn(agg, stats, bn1g, bn1b, agg, stream);                       // h1 in agg

  // --- Mamba ---
  launch_gemm(x, inpw, nullptr, xz, N_NODES, 2 * CH, CH, 0, stream); // [x_in | z]
  k_conv_silu<<<ncBlocks, EW, 0, stream>>>(xz, convw, convb, xc);
  k_pad_xproj<<<(CH * XPN + EW - 1) / EW, EW, 0, stream>>>(xprojw, xpw);
  launch_gemm(xc, xpw, nullptr, dbl, N_NODES, XPN, CH, 0, stream);   // [dt8 | B16 | C16 | pad]
  k_dt<<<ncBlocks, EW, 0, stream>>>(dbl, dtw, dtb, dtb_);
  k_scan<<<N_GRAPHS, CH, 0, stream>>>(dbl, dtb_, xc, alog, y);
  k_gate<<<ncBlocks, EW, 0, stream>>>(xz, xc, Dp, y);
  launch_gemm(y, outpw, nullptr, mo, N_NODES, CH, CH, 0, stream);    // mamba out
  k_add<<<ncBlocks, EW, 0, stream>>>(mo, x, mo, NC);                 // + residual
  run_bn(mo, stats, bn2g, bn2b, mo, stream);                         // h2 in mo

  // --- combine + MLP ---
  k_add<<<ncBlocks, EW, 0, stream>>>(agg, mo, xw, NC);               // out0 in xw
  launch_gemm(xw, w1, b1, xz, N_NODES, 2 * CH, CH, 1, stream);       // relu hidden
  launch_gemm(xz, w2, b2, xc, N_NODES, CH, 2 * CH, 0, stream);       // mlp out
  k_add<<<ncBlocks, EW, 0, stream>>>(xw, xc, xc, NC);                // + residual
  run_bn(xc, stats, bn3g, bn3b, out, stream);                        // final BN -> d_out
}